// QuantumNATGen257_65481071397730
// MI455X (gfx1250) — compile-verified
//
#include <hip/hip_runtime.h>
#include <math.h>

// ---------------------------------------------------------------------------
// QuantumNAT forward for MI455X (gfx1250, wave32, WMMA).
//
// z[b] = e_b^T (V^H diag(sign_w) V) e_b  collapses to:
//   re = Re(V)*e ; im = Im(V)*e ; p = re^2+im^2 ; z_w = sum_i sign_w(i) p_i
// with e_b a REAL 16-vector (tensor product of per-wire (cos,sin) halves) and
// V the fixed 16x16 circuit unitary built once per block from `weights`.
// The 16x16x16 real matvec batches map onto V_WMMA_F32_16X16X4_F32 (K chained
// 4x): 8 fp32 WMMAs per 16-sample tile -> full fp32 precision for the
// batch-norm tolerance; overall memory-bound (~64 MB total traffic @23.3TB/s).
//
// Round-1 fix: B fragments are now formed with purely static register
// indexing (the lane-dependent K offset collapses to a single c2/s2 select),
// eliminating the v_cmp/v_cndmask select trees clang generated for the
// dynamically-indexed e[16] array. Encoder uses hw v_sin/v_cos.
// ---------------------------------------------------------------------------

typedef __attribute__((ext_vector_type(2))) float v2f;
typedef __attribute__((ext_vector_type(8))) float v8f;

#define QN_BATCH   (1 << 20)
#define QN_BLOCKS  1024
#define QN_THREADS 256       // 8 waves/block; 8 iters x 8 waves x 16 = 1024 samples/block
#define QN_EPS     1e-5f

__global__ __launch_bounds__(QN_THREADS)
void qnat_fwd(const float* __restrict__ x, const float* __restrict__ wts,
              float* __restrict__ zout, float* __restrict__ partials)
{
    __shared__ float sVre[256];
    __shared__ float sVim[256];
    __shared__ float sW[64];          // per-wave partial sums [8 waves][8]

    const int tid = threadIdx.x;

    // ---- build fixed circuit unitary V: thread j evolves basis column e_j ----
    if (tid < 16) {
        float sre[16], sim[16];
        #pragma unroll
        for (int i = 0; i < 16; ++i) { sre[i] = (i == tid) ? 1.f : 0.f; sim[i] = 0.f; }
        #pragma unroll
        for (int l = 0; l < 2; ++l) {
            #pragma unroll
            for (int w = 0; w < 4; ++w) {
                const float phi = wts[(l * 4 + w) * 3 + 0];
                const float the = wts[(l * 4 + w) * 3 + 1];
                const float omg = wts[(l * 4 + w) * 3 + 2];
                float st_, ct;  sincosf(0.5f * the, &st_, &ct);          // precise: once/block
                float sp, cp;   sincosf(0.5f * (phi + omg), &sp, &cp);   // ep = cp - i sp
                float sm, cm;   sincosf(0.5f * (phi - omg), &sm, &cm);   // em = cm + i sm
                // Rot = [[ep*c, -em*s], [conj(em)*s, conj(ep)*c]]
                const float u00r =  cp * ct,  u00i = -sp * ct;
                const float u01r = -cm * st_, u01i = -sm * st_;
                const float u10r =  cm * st_, u10i = -sm * st_;
                const float u11r =  cp * ct,  u11i =  sp * ct;
                const int mask = 1 << (3 - w);   // wire 0 is the MSB
                #pragma unroll
                for (int i = 0; i < 16; ++i) {
                    if (!(i & mask)) {
                        const int j = i | mask;
                        const float ar = sre[i], ai = sim[i];
                        const float br = sre[j], bi = sim[j];
                        sre[i] = u00r*ar - u00i*ai + u01r*br - u01i*bi;
                        sim[i] = u00r*ai + u00i*ar + u01r*bi + u01i*br;
                        sre[j] = u10r*ar - u10i*ai + u11r*br - u11i*bi;
                        sim[j] = u10r*ai + u10i*ar + u11r*bi + u11i*br;
                    }
                }
            }
            #pragma unroll
            for (int c = 0; c < 3; ++c) {       // CNOT ladder, control c, target c+1
                const int cmask = 1 << (3 - c);
                const int tmask = 1 << (2 - c);
                #pragma unroll
                for (int i = 0; i < 16; ++i) {
                    if ((i & cmask) && !(i & tmask)) {
                        const int j = i | tmask;
                        float t = sre[i]; sre[i] = sre[j]; sre[j] = t;
                        t = sim[i]; sim[i] = sim[j]; sim[j] = t;
                    }
                }
            }
        }
        #pragma unroll
        for (int i = 0; i < 16; ++i) { sVre[i * 16 + tid] = sre[i]; sVim[i * 16 + tid] = sim[i]; }
    }
    __syncthreads();

    const int lane = tid & 31;
    const int wave = tid >> 5;
    const int m    = lane & 15;
    const int koff = (lane & 16) ? 2 : 0;
    const bool hi  = (lane & 16) != 0;

    // A fragments from LDS (ISA 7.12.2: A lane = row M; VGPR0/1 = K{0,1} lo-half,
    // K{2,3} hi-half). Runtime LDS addressing is fine (ds_load, no select trees).
    v2f aRe[4], aIm[4];
    #pragma unroll
    for (int kc = 0; kc < 4; ++kc) {
        aRe[kc].x = sVre[m * 16 + 4 * kc + koff];
        aRe[kc].y = sVre[m * 16 + 4 * kc + koff + 1];
        aIm[kc].x = sVim[m * 16 + 4 * kc + koff];
        aIm[kc].y = sVim[m * 16 + 4 * kc + koff + 1];
    }

    float accS[4] = {0.f, 0.f, 0.f, 0.f};
    float accQ[4] = {0.f, 0.f, 0.f, 0.f};
    const float4* __restrict__ x4 = (const float4*)x;

    #pragma unroll 1
    for (int it = 0; it < 8; ++it) {
        const int tile = blockIdx.x * 64 + it * 8 + wave;
        const int s    = tile * 16 + m;

        // ---- encoder: e_j = g01[j>>2] * g23[j&3]. For B-chunk kc the needed
        // indices are j = 4kc+koff(+1): j>>2 == kc (static), j&3 == koff(+1),
        // and g23[koff]/g23[koff+1] = (hi ? s2 : c2) * {c3, s3}. One select.
        const float4 xv = x4[s];
        float s0, c0, s1, c1, s2, c2, s3, c3;
        __sincosf(0.5f * xv.x, &s0, &c0);       // hw v_sin/v_cos: |err| ~1e-6,
        __sincosf(0.5f * xv.y, &s1, &c1);       // negligible after batch-norm
        __sincosf(0.5f * xv.z, &s2, &c2);
        __sincosf(0.5f * xv.w, &s3, &c3);
        const float g01_0 = c0 * c1, g01_1 = c0 * s1, g01_2 = s0 * c1, g01_3 = s0 * s1;
        const float h2 = hi ? s2 : c2;
        const float t0 = h2 * c3;
        const float t1 = h2 * s3;

        // ---- amp = V * e : two fp32 WMMA chains (K = 16 as 4 x K=4) ----
        v2f b0; b0.x = g01_0 * t0; b0.y = g01_0 * t1;
        v2f b1; b1.x = g01_1 * t0; b1.y = g01_1 * t1;
        v2f b2; b2.x = g01_2 * t0; b2.y = g01_2 * t1;
        v2f b3; b3.x = g01_3 * t0; b3.y = g01_3 * t1;

        v8f re = {}; v8f im = {};
        re = __builtin_amdgcn_wmma_f32_16x16x4_f32(false, aRe[0], false, b0, (short)0, re, false, false);
        im = __builtin_amdgcn_wmma_f32_16x16x4_f32(false, aIm[0], false, b0, (short)0, im, false, false);
        re = __builtin_amdgcn_wmma_f32_16x16x4_f32(false, aRe[1], false, b1, (short)0, re, false, false);
        im = __builtin_amdgcn_wmma_f32_16x16x4_f32(false, aIm[1], false, b1, (short)0, im, false, false);
        re = __builtin_amdgcn_wmma_f32_16x16x4_f32(false, aRe[2], false, b2, (short)0, re, false, false);
        im = __builtin_amdgcn_wmma_f32_16x16x4_f32(false, aIm[2], false, b2, (short)0, im, false, false);
        re = __builtin_amdgcn_wmma_f32_16x16x4_f32(false, aRe[3], false, b3, (short)0, re, false, false);
        im = __builtin_amdgcn_wmma_f32_16x16x4_f32(false, aIm[3], false, b3, (short)0, im, false, false);

        // ---- probabilities + signed sums (lane holds amps r(+8*hi) of sample m)
        float p[8];
        #pragma unroll
        for (int r = 0; r < 8; ++r) p[r] = re[r] * re[r] + im[r] * im[r];

        const float s0123 = (p[0] + p[1]) + (p[2] + p[3]);
        const float s4567 = (p[4] + p[5]) + (p[6] + p[7]);
        float z3 = (p[0] - p[1]) + (p[2] - p[3]) + (p[4] - p[5]) + (p[6] - p[7]);
        float z2 = (p[0] + p[1]) - (p[2] + p[3]) + (p[4] + p[5]) - (p[6] + p[7]);
        float z1 = s0123 - s4567;
        float z0 = hi ? -(s0123 + s4567) : (s0123 + s4567);

        // combine the two half-waves (amps 0-7 live in lane, 8-15 in lane^16)
        z0 += __shfl_xor(z0, 16, 32);
        z1 += __shfl_xor(z1, 16, 32);
        z2 += __shfl_xor(z2, 16, 32);
        z3 += __shfl_xor(z3, 16, 32);

        if (!hi) {
            ((float4*)zout)[s] = make_float4(z0, z1, z2, z3);
            accS[0] += z0; accS[1] += z1; accS[2] += z2; accS[3] += z3;
            accQ[0] += z0 * z0; accQ[1] += z1 * z1;
            accQ[2] += z2 * z2; accQ[3] += z3 * z3;
        }
    }

    // ---- deterministic block reduction: fixed shuffle tree, then 8-way serial add
    float acc8[8] = { accS[0], accS[1], accS[2], accS[3],
                      accQ[0], accQ[1], accQ[2], accQ[3] };
    #pragma unroll
    for (int k = 0; k < 8; ++k) {
        #pragma unroll
        for (int off = 16; off >= 1; off >>= 1)
            acc8[k] += __shfl_xor(acc8[k], off, 32);
    }
    if (lane == 0) {
        #pragma unroll
        for (int k = 0; k < 8; ++k) sW[wave * 8 + k] = acc8[k];
    }
    __syncthreads();
    if (tid < 8) {
        float t = 0.f;
        #pragma unroll
        for (int wv = 0; wv < 8; ++wv) t += sW[wv * 8 + tid];
        partials[blockIdx.x * 8 + tid] = t;
    }
}

__global__ void qnat_stats(const float* __restrict__ partials,
                           const float* __restrict__ gamma,
                           const float* __restrict__ beta,
                           float* __restrict__ stats)
{
    __shared__ float s[8];
    const int t = threadIdx.x;
    if (t < 8) {
        double tot = 0.0;                       // deterministic fixed-order sum
        for (int b = 0; b < QN_BLOCKS; ++b) tot += (double)partials[b * 8 + t];
        s[t] = (float)tot;
    }
    __syncthreads();
    if (t < 4) {
        const float invB = 1.f / (float)QN_BATCH;
        const float mean = s[t] * invB;
        const float var  = s[4 + t] * invB - mean * mean;   // biased variance
        const float rstd = 1.f / sqrtf(var + QN_EPS);
        const float sc   = gamma[t] * rstd;
        stats[t]     = sc;
        stats[4 + t] = beta[t] - mean * sc;
    }
}

__global__ __launch_bounds__(256)
void qnat_bn(float* __restrict__ zout, const float* __restrict__ stats)
{
    const int i = blockIdx.x * blockDim.x + threadIdx.x;
    float4* z4 = (float4*)zout;
    float4 z = z4[i];
    z.x = z.x * stats[0] + stats[4];
    z.y = z.y * stats[1] + stats[5];
    z.z = z.z * stats[2] + stats[6];
    z.w = z.w * stats[3] + stats[7];
    z4[i] = z;
}

extern "C" void kernel_launch(void* const* d_in, const int* in_sizes, int n_in,
                              void* d_out, int out_size, void* d_ws, size_t ws_size,
                              hipStream_t stream)
{
    const float* x  = (const float*)d_in[0];   // (B, 4)
    const float* wt = (const float*)d_in[1];   // (2, 4, 3)
    const float* gm = (const float*)d_in[2];   // (4,)
    const float* bt = (const float*)d_in[3];   // (4,)
    float* z        = (float*)d_out;           // (B, 4) — z in pass 1, normalized in pass 3
    float* partials = (float*)d_ws;            // QN_BLOCKS * 8 floats
    float* stats    = partials + QN_BLOCKS * 8;

    qnat_fwd  <<<QN_BLOCKS, QN_THREADS, 0, stream>>>(x, wt, z, partials);
    qnat_stats<<<1, 32, 0, stream>>>(partials, gm, bt, stats);
    qnat_bn   <<<QN_BATCH / 256, 256, 0, stream>>>(z, stats);
}